// SparseGCN_72756745994565
// MI455X (gfx1250) — compile-verified
//
#include <hip/hip_runtime.h>

// ---------------------------------------------------------------------------
// SparseGCN for MI455X (gfx1250):
//   * fp32 WMMA GEMMs (register-resident B fragments, K fully unrolled)
//   * CSR-by-dst built on device -> atomic-free gather aggregation
//   * self-loop + bias + LeakyReLU fused into the gather kernel
// ---------------------------------------------------------------------------

typedef __attribute__((ext_vector_type(2))) float v2f;
typedef __attribute__((ext_vector_type(8))) float v8f;

constexpr int NN    = 80000;    // nodes
constexpr int EE    = 1280000;  // edges
constexpr int F_IN  = 128;
constexpr int HH    = 64;
constexpr int OUTF  = 32;
constexpr int NLAYR = 3;
constexpr float SELF_W = 2.0f;
constexpr float SLOPE  = 0.01f;

constexpr int SCAN_T  = 256;
constexpr int SCAN_B  = (NN + SCAN_T - 1) / SCAN_T;   // 313

// ---------------------------------------------------------------------------
// CSR construction: histogram -> 2-level exclusive scan -> fill
// ---------------------------------------------------------------------------
__global__ void gcn_init_counts(int* __restrict__ cnt, int* __restrict__ fill) {
    int i = blockIdx.x * blockDim.x + threadIdx.x;
    if (i < NN) { cnt[i] = 0; fill[i] = 0; }
}

__global__ void gcn_accum_counts(const int* __restrict__ dst, int* __restrict__ cnt) {
    int e = blockIdx.x * blockDim.x + threadIdx.x;
    if (e < EE) atomicAdd(&cnt[dst[e]], 1);
}

// dis = rsqrt(cnt + 2), self_norm = 2 * dis^2
__global__ void gcn_compute_dis(const int* __restrict__ cnt,
                                float* __restrict__ dis,
                                float* __restrict__ self_norm) {
    int i = blockIdx.x * blockDim.x + threadIdx.x;
    if (i < NN) {
        float d = __frsqrt_rn((float)cnt[i] + SELF_W);
        dis[i] = d;
        self_norm[i] = SELF_W * d * d;
    }
}

// per-block exclusive scan; emits block totals
__global__ void gcn_scan_blocks(const int* __restrict__ cnt,
                                int* __restrict__ excl,
                                int* __restrict__ blockSums) {
    __shared__ int sh[SCAN_T];
    int i = blockIdx.x * SCAN_T + threadIdx.x;
    int v = (i < NN) ? cnt[i] : 0;
    sh[threadIdx.x] = v;
    __syncthreads();
#pragma unroll
    for (int off = 1; off < SCAN_T; off <<= 1) {
        int t = (threadIdx.x >= off) ? sh[threadIdx.x - off] : 0;
        __syncthreads();
        sh[threadIdx.x] += t;
        __syncthreads();
    }
    if (i < NN) excl[i] = sh[threadIdx.x] - v;            // exclusive
    if (threadIdx.x == SCAN_T - 1) blockSums[blockIdx.x] = sh[SCAN_T - 1];
}

// single-block exclusive scan of the block sums (313 <= 512)
__global__ void gcn_scan_sums(int* __restrict__ blockSums) {
    __shared__ int sh[512];
    int v = (threadIdx.x < SCAN_B) ? blockSums[threadIdx.x] : 0;
    sh[threadIdx.x] = v;
    __syncthreads();
#pragma unroll
    for (int off = 1; off < 512; off <<= 1) {
        int t = (threadIdx.x >= off) ? sh[threadIdx.x - off] : 0;
        __syncthreads();
        sh[threadIdx.x] += t;
        __syncthreads();
    }
    if (threadIdx.x < SCAN_B) blockSums[threadIdx.x] = sh[threadIdx.x] - v;
}

__global__ void gcn_scan_add(int* __restrict__ excl, const int* __restrict__ blockSums) {
    int i = blockIdx.x * SCAN_T + threadIdx.x;
    if (i < NN) excl[i] += blockSums[blockIdx.x];
}

// csr_src[pos] = src ; csr_w[pos] = dis[src]*dis[dst]
__global__ void gcn_csr_fill(const int* __restrict__ src, const int* __restrict__ dst,
                             const float* __restrict__ dis,
                             const int* __restrict__ rowptr, int* __restrict__ fill,
                             int* __restrict__ csr_src, float* __restrict__ csr_w) {
    int e = blockIdx.x * blockDim.x + threadIdx.x;
    if (e < EE) {
        int s = src[e], d = dst[e];
        int pos = rowptr[d] + atomicAdd(&fill[d], 1);
        csr_src[pos] = s;
        csr_w[pos] = dis[s] * dis[d];
    }
}

// ---------------------------------------------------------------------------
// WMMA fp32 GEMM: C[M,N] = A[M,K] @ B[K,N] (+ bias)
//   one wave -> one 16-col strip; B fragments live in registers for the whole
//   kernel; grid-stride over 16-row tiles amortizes the B load.
// V_WMMA_F32_16X16X4_F32 layout (ISA 7.12.2):
//   A (16x4): lane<16 -> M=lane, v0=K0 v1=K1 ; lane>=16 -> M=lane-16, v0=K2 v1=K3
//   B (4x16): same striping with N=lane%16
//   C (16x16): VGPR r -> M = r + 8*(lane>=16), N = lane%16
// ---------------------------------------------------------------------------
template <int K>
__global__ void gcn_wmma_gemm(const float* __restrict__ A,
                              const float* __restrict__ B,
                              const float* __restrict__ bias,   // may be null
                              float* __restrict__ C,
                              int N, int mTiles) {
    constexpr int S = K / 4;
    const int lane = threadIdx.x & 31;
    const int wave = threadIdx.x >> 5;
    const int half = lane >> 4;        // K sub-pair select
    const int lr   = lane & 15;
    const int col  = wave * 16 + lr;

    // Register-resident B fragments (<= 64 VGPRs for K=128)
    v2f bf[S];
#pragma unroll
    for (int s = 0; s < S; ++s) {
        bf[s].x = B[(size_t)(4 * s + half * 2)     * N + col];
        bf[s].y = B[(size_t)(4 * s + half * 2 + 1) * N + col];
    }
    const float bv = bias ? bias[col] : 0.0f;

    for (int rt = blockIdx.x; rt < mTiles; rt += gridDim.x) {
        const float* Arow = A + (size_t)(rt * 16 + lr) * K + half * 2;
        v8f acc = {};
#pragma unroll
        for (int s = 0; s < S; ++s) {
            v2f a;
            a.x = Arow[4 * s];
            a.y = Arow[4 * s + 1];
            acc = __builtin_amdgcn_wmma_f32_16x16x4_f32(
                false, a, false, bf[s], (short)0, acc, false, false);
        }
        float* Crow = C + (size_t)(rt * 16 + half * 8) * N + col;
#pragma unroll
        for (int r = 0; r < 8; ++r)
            Crow[(size_t)r * N] = acc[r] + bv;
    }
}

// ---------------------------------------------------------------------------
// Fused aggregation: one wave per node, 2 features per lane (H=64).
//   h[n] = leaky( g[n]*self_norm[n] + bias + sum_in-edges g[src]*w )
// Atomic-free; outputs written exactly once. Software-pipelined CSR reads
// with a gfx1250 global_prefetch of the next neighbor's feature row.
// ---------------------------------------------------------------------------
__global__ void gcn_gather(const float* __restrict__ g,
                           const int* __restrict__ rowptr,
                           const int* __restrict__ cnt,
                           const int* __restrict__ csr_src,
                           const float* __restrict__ csr_w,
                           const float* __restrict__ self_norm,
                           const float* __restrict__ bias,
                           float* __restrict__ h) {
    int n = (int)((blockIdx.x * (long)blockDim.x + threadIdx.x) >> 5);
    int lane = threadIdx.x & 31;
    if (n >= NN) return;
    const int f2 = lane * 2;

    float2 v = *(const float2*)(g + (size_t)n * HH + f2);
    const float sn = self_norm[n];
    float ax = v.x * sn + bias[f2];
    float ay = v.y * sn + bias[f2 + 1];

    const int beg = rowptr[n];
    const int len = cnt[n];

    if (len > 0) {
        int   sNext = csr_src[beg];
        float wNext = csr_w[beg];
        for (int j = 0; j < len; ++j) {
            const int   s = sNext;
            const float w = wNext;
            const float* grow = g + (size_t)s * HH + f2;
            if (j + 1 < len) {
                sNext = csr_src[beg + j + 1];
                wNext = csr_w[beg + j + 1];
                __builtin_prefetch(g + (size_t)sNext * HH + f2, 0, 0);
            }
            const float2 u = *(const float2*)grow;
            ax += u.x * w;
            ay += u.y * w;
        }
    }

    float2 r;
    r.x = fmaxf(ax, SLOPE * ax);     // LeakyReLU (slope < 1)
    r.y = fmaxf(ay, SLOPE * ay);
    *(float2*)(h + (size_t)n * HH + f2) = r;
}

// ---------------------------------------------------------------------------
// Host-side orchestration
// ---------------------------------------------------------------------------
extern "C" void kernel_launch(void* const* d_in, const int* in_sizes, int n_in,
                              void* d_out, int out_size, void* d_ws, size_t ws_size,
                              hipStream_t stream) {
    (void)in_sizes; (void)n_in; (void)out_size; (void)ws_size;

    const float* x      = (const float*)d_in[0];   // [N, F_IN]
    const int*   eidx   = (const int*)d_in[1];     // [2, E]
    const float* W_pre  = (const float*)d_in[2];   // [F_IN, H]
    const float* b_pre  = (const float*)d_in[3];   // [H]
    const float* Ws     = (const float*)d_in[4];   // [L, H, H]
    const float* bs     = (const float*)d_in[5];   // [L, H]
    const float* W_post = (const float*)d_in[6];   // [H, OUT]
    const float* b_post = (const float*)d_in[7];   // [OUT]
    float*       out    = (float*)d_out;           // [N, OUT]

    const int* src = eidx;
    const int* dst = eidx + EE;

    // Carve workspace (256B aligned slices)
    char* wsp = (char*)d_ws;
    auto carve = [&](size_t bytes) -> void* {
        void* p = (void*)wsp;
        wsp += (bytes + 255) & ~(size_t)255;
        return p;
    };
    int*   cnt       = (int*)  carve((size_t)NN * 4);
    int*   fill      = (int*)  carve((size_t)NN * 4);
    int*   rowptr    = (int*)  carve((size_t)NN * 4);
    int*   blockSums = (int*)  carve((size_t)512 * 4);
    int*   csr_src   = (int*)  carve((size_t)EE * 4);
    float* csr_w     = (float*)carve((size_t)EE * 4);
    float* dis       = (float*)carve((size_t)NN * 4);
    float* self_norm = (float*)carve((size_t)NN * 4);
    float* h         = (float*)carve((size_t)NN * HH * 4);
    float* g         = (float*)carve((size_t)NN * HH * 4);

    const int T  = 256;
    const int bN = (NN + T - 1) / T;
    const int bE = (EE + T - 1) / T;
    const int bG = (NN * 32 + T - 1) / T;   // gather: 1 wave per node
    const int mTiles = NN / 16;             // 5000
    const int gemmGrid = 512;               // grid-stride over row tiles

    // 1) CSR by destination + normalization coefficients
    gcn_init_counts<<<bN, T, 0, stream>>>(cnt, fill);
    gcn_accum_counts<<<bE, T, 0, stream>>>(dst, cnt);
    gcn_compute_dis<<<bN, T, 0, stream>>>(cnt, dis, self_norm);
    gcn_scan_blocks<<<SCAN_B, SCAN_T, 0, stream>>>(cnt, rowptr, blockSums);
    gcn_scan_sums<<<1, 512, 0, stream>>>(blockSums);
    gcn_scan_add<<<SCAN_B, SCAN_T, 0, stream>>>(rowptr, blockSums);
    gcn_csr_fill<<<bE, T, 0, stream>>>(src, dst, dis, rowptr, fill, csr_src, csr_w);

    // 2) h = x @ W_pre + b_pre   (K=128, N=64 -> 4 waves/block)
    gcn_wmma_gemm<F_IN><<<gemmGrid, 4 * 32, 0, stream>>>(x, W_pre, b_pre, h, HH, mTiles);

    // 3) GCN layers: GEMM then fused gather/self-loop/bias/LeakyReLU
    for (int l = 0; l < NLAYR; ++l) {
        const float* Wl = Ws + (size_t)l * HH * HH;
        const float* bl = bs + (size_t)l * HH;
        gcn_wmma_gemm<HH><<<gemmGrid, 4 * 32, 0, stream>>>(h, Wl, nullptr, g, HH, mTiles);
        gcn_gather<<<bG, T, 0, stream>>>(g, rowptr, cnt, csr_src, csr_w,
                                         self_norm, bl, h);
    }

    // 4) out = h @ W_post + b_post   (K=64, N=32 -> 2 waves/block)
    gcn_wmma_gemm<HH><<<gemmGrid, 2 * 32, 0, stream>>>(h, W_post, b_post, out, HH, mTiles);
}